// FESTGCN_83021717831859
// MI455X (gfx1250) — compile-verified
//
#include <hip/hip_runtime.h>
#include <math.h>

#define B_  4
#define T_  10
#define N_  4096
#define H_  32

typedef __attribute__((ext_vector_type(2))) float v2f;
typedef __attribute__((ext_vector_type(8))) float v8f;

__device__ __forceinline__ float sigmoidf_(float x) { return 1.0f / (1.0f + __expf(-x)); }

// ---------------------------------------------------------------------------
// Zero-init workspace accumulators (gcn1, gcn2)
// ---------------------------------------------------------------------------
__global__ __launch_bounds__(256) void zero_ws(float* __restrict__ p, int n) {
    int i = blockIdx.x * 256 + threadIdx.x;
    if (i < n) p[i] = 0.0f;
}

// ---------------------------------------------------------------------------
// Z = conc @ W  (per node, [H+1] -> [KH]); optionally gates h by sigmoid(gcn1)
// Z layout: [N, B*KH], column c = b*KH + k
// one block per (b,n), blockDim = KH (64 for W1, 32 for W2)
// ---------------------------------------------------------------------------
__global__ __launch_bounds__(64) void prep_kernel(
    const float* __restrict__ inputs,   // [B,T,N]
    const float* __restrict__ states,   // [T,B,N*H]
    const float* __restrict__ W,        // [H+1, KH] row-major
    float* __restrict__ Z,              // [N, B*KH]
    const float* __restrict__ gcn1,     // [B, N*2H] (flat), used iff useGate
    int t, int KH, int useGate)
{
    const int blk = blockIdx.x;         // 0 .. B*N-1
    const int b   = blk >> 12;          // / N_ (4096)
    const int n   = blk & (N_ - 1);
    const int k   = threadIdx.x;        // 0..KH-1

    __shared__ float hs[H_];
    if (k < H_) {
        float h = states[((size_t)t * B_ + b) * (size_t)(N_ * H_) + (size_t)n * H_ + k];
        if (useGate) {
            // r = sigmoid(first half of flattened gcn1), flat index n*H + k
            float g = gcn1[(size_t)b * (N_ * 2 * H_) + (size_t)n * H_ + k];
            h *= sigmoidf_(g);
        }
        hs[k] = h;
    }
    __syncthreads();

    const float x = inputs[((size_t)b * T_ + t) * N_ + n];
    float acc = x * W[k];               // W[0][k]
    #pragma unroll
    for (int j = 0; j < H_; ++j)
        acc += hs[j] * W[(j + 1) * KH + k];

    Z[(size_t)n * (B_ * KH) + b * KH + k] = acc;
}

// ---------------------------------------------------------------------------
// gcn += A_t @ Z + bias   using V_WMMA_F32_16X16X4_F32
//   - A_t built on the fly from dtw/spec_lap/time_delay (+eye, +laplacian last
//     step); dtw+spec+td = 192 MB = L2-resident across all GEMMs.
//   - Z tile copied global->LDS with GLOBAL_LOAD_ASYNC_TO_LDS_B128 (ASYNCcnt),
//     overlapping the DMA with A-tile construction.
// Block: 256 threads (8 wave32), tile 64 rows x 64 cols, K-chunk 32.
// Wave (rw = wid&3, cw = wid>>2): 16 rows x 32 cols = 2 accumulator tiles.
// ---------------------------------------------------------------------------
#define BS_STRIDE 68   // 272 B row stride: 16B-aligned for B128 LDS writes

__global__ __launch_bounds__(256) void gemm_wmma_acc(
    const float* __restrict__ dtw,
    const float* __restrict__ spec,
    const float* __restrict__ tdl,
    const float* __restrict__ lap,
    const float* __restrict__ Z,        // [N_, colsTotal]
    float*       __restrict__ gcn,      // [B, N_, KH] flat, accumulated
    const float* __restrict__ bias,     // [KH]
    int t, int isLast, int KH, int khShift, int colsTotal)
{
    const float scale = isLast ? (1.0f / 3.0f) : 0.5f;
    const float tthr  = (float)(t + 1);
    const int m0 = blockIdx.x * 64;
    const int c0 = blockIdx.y * 64;

    __shared__ float As[64][33];                        // +1 pad: conflict-free column reads
    __shared__ __align__(16) float Bs[32][BS_STRIDE];   // async-DMA destination

    const int tid  = threadIdx.x;
    const int wid  = tid >> 5;
    const int lane = tid & 31;
    const int lr   = lane & 15;
    const int kh   = (lane >> 4) << 1;      // 0 or 2: K-pair owned by this lane half
    const int rw   = wid & 3;               // row-group (x16)
    const int cw   = wid >> 2;              // col-group (x32)

    // LDS byte offset of Bs: low 32 bits of the flat address (LDS aperture
    // lives in the high 32 bits, so addr[31:0] == LDS offset per ISA).
    const unsigned bsOff = (unsigned)(unsigned long long)(const void*)&Bs[0][0];

    v8f acc[2] = {};

    for (int kb = 0; kb < N_; kb += 32) {
        // ---- kick off async DMA of Z tile: 32 rows x 64 cols, 2 B128/thread ----
        #pragma unroll
        for (int it = 0; it < 2; ++it) {
            int i   = it * 256 + tid;           // 0..511 B128 transfers
            int row = i >> 4;                   // 0..31
            int c4  = (i & 15) << 2;            // 0,4,...,60 (floats)
            unsigned gOff = (unsigned)(((kb + row) * colsTotal + c0 + c4) * 4);
            unsigned lOff = bsOff + (unsigned)((row * BS_STRIDE + c4) * 4);
            asm volatile("global_load_async_to_lds_b128 %0, %1, %2"
                         :: "v"(lOff), "v"(gOff), "s"(Z) : "memory");
        }

        // ---- build A tile (fused mask/eye/scale), coalesced along K (j) ----
        #pragma unroll
        for (int it = 0; it < 8; ++it) {
            int idx = it * 256 + tid;
            int row = idx >> 5, col = idx & 31;
            int m = m0 + row;
            int j = kb + col;
            size_t o = (size_t)m * N_ + j;
            float d   = dtw[o];
            float tdv = ceilf(fabsf(tdl[o]));
            float dm  = ((d != 0.0f) && (tthr > (float)T_ - tdv)) ? d : 0.0f;
            float a   = dm + spec[o] + ((m == j) ? 1.0f : 0.0f);
            if (isLast) a += lap[o];
            As[row][col] = a * scale;
        }

        // ---- wait for the DMA, then converge the block ----
        asm volatile("s_wait_asynccnt 0x0" ::: "memory");
        __syncthreads();

        const int mrow = rw * 16 + lr;
        #pragma unroll
        for (int kk = 0; kk < 32; kk += 4) {
            // A frag 16x4: lane = row; v0/v1 = K {kh, kh+1}
            v2f a;
            a.x = As[mrow][kk + kh];
            a.y = As[mrow][kk + kh + 1];
            #pragma unroll
            for (int ct = 0; ct < 2; ++ct) {
                // B frag 4x16: lane = col; v0/v1 = K {kh, kh+1}
                v2f bf;
                bf.x = Bs[kk + kh][cw * 32 + ct * 16 + lr];
                bf.y = Bs[kk + kh + 1][cw * 32 + ct * 16 + lr];
                acc[ct] = __builtin_amdgcn_wmma_f32_16x16x4_f32(
                    false, a, false, bf, (short)0, acc[ct], false, false);
            }
        }
        __syncthreads();
    }

    // ---- epilogue: gcn[b, row, k] += acc + bias[k] (unique writer/element) ----
    const int rowBase = m0 + rw * 16 + ((lane >> 4) << 3);  // C: M = i (+8 for hi lanes)
    #pragma unroll
    for (int ct = 0; ct < 2; ++ct) {
        int ccol = c0 + cw * 32 + ct * 16 + lr;
        int b = ccol >> khShift;
        int k = ccol & (KH - 1);
        float bv = bias[k];
        #pragma unroll
        for (int i = 0; i < 8; ++i) {
            int row = rowBase + i;
            size_t o = ((size_t)b * N_ + row) * KH + k;
            gcn[o] += acc[ct][i] + bv;
        }
    }
}

// ---------------------------------------------------------------------------
// h_new = u*h + (1-u)*tanh(gcn2);  u = sigmoid(second half of flat gcn1)
// ---------------------------------------------------------------------------
__global__ __launch_bounds__(256) void final_kernel(
    const float* __restrict__ states,
    const float* __restrict__ gcn1,
    const float* __restrict__ gcn2,
    float* __restrict__ out)
{
    int i = blockIdx.x * 256 + threadIdx.x;
    if (i >= B_ * N_ * H_) return;
    int b   = i / (N_ * H_);
    int loc = i - b * (N_ * H_);
    float u = sigmoidf_(gcn1[(size_t)b * (N_ * 2 * H_) + (N_ * H_) + loc]);
    float h = states[((size_t)(T_ - 1) * B_ + b) * (size_t)(N_ * H_) + loc];
    float c = tanhf(gcn2[i]);
    out[i] = u * h + (1.0f - u) * c;
}

// ---------------------------------------------------------------------------
extern "C" void kernel_launch(void* const* d_in, const int* in_sizes, int n_in,
                              void* d_out, int out_size, void* d_ws, size_t ws_size,
                              hipStream_t stream) {
    const float* inputs = (const float*)d_in[0];
    const float* states = (const float*)d_in[1];
    const float* dtw    = (const float*)d_in[2];
    const float* spec   = (const float*)d_in[3];
    const float* lap    = (const float*)d_in[4];
    const float* tdl    = (const float*)d_in[5];
    const float* W1     = (const float*)d_in[6];
    const float* b1     = (const float*)d_in[7];
    const float* W2     = (const float*)d_in[8];
    const float* b2     = (const float*)d_in[9];
    float* out = (float*)d_out;

    float* ws   = (float*)d_ws;
    float* gcn1 = ws;                                   // B*N*2H = 1,048,576 f32
    float* gcn2 = gcn1 + (size_t)B_ * N_ * 2 * H_;      // B*N*H  =   524,288 f32
    float* Z1   = gcn2 + (size_t)B_ * N_ * H_;          // N*256  = 1,048,576 f32
    float* Z2   = Z1   + (size_t)N_ * (B_ * 2 * H_);    // N*128  =   524,288 f32
    // total ~12.5 MB of workspace

    const int nz = B_ * N_ * 2 * H_ + B_ * N_ * H_;     // gcn1 + gcn2 contiguous
    zero_ws<<<(nz + 255) / 256, 256, 0, stream>>>(gcn1, nz);

    for (int t = 0; t < T_; ++t) {
        const int isLast = (t == T_ - 1) ? 1 : 0;

        // Z1 = conc1 @ W1 ; conc1 = [x_t, h_t]
        prep_kernel<<<B_ * N_, 2 * H_, 0, stream>>>(
            inputs, states, W1, Z1, gcn1, t, 2 * H_, /*useGate=*/0);

        // gcn1 += A_t @ Z1 + b1   (cols = B*2H = 256)
        gemm_wmma_acc<<<dim3(N_ / 64, (B_ * 2 * H_) / 64), 256, 0, stream>>>(
            dtw, spec, tdl, lap, Z1, gcn1, b1, t, isLast,
            /*KH=*/2 * H_, /*khShift=*/6, /*colsTotal=*/B_ * 2 * H_);

        // Z2 = conc2 @ W2 ; conc2 = [x_t, sigmoid(r)*h_t]  (flat-chunk gate)
        prep_kernel<<<B_ * N_, H_, 0, stream>>>(
            inputs, states, W2, Z2, gcn1, t, H_, /*useGate=*/1);

        // gcn2 += A_t @ Z2 + b2   (cols = B*H = 128)
        gemm_wmma_acc<<<dim3(N_ / 64, (B_ * H_) / 64), 256, 0, stream>>>(
            dtw, spec, tdl, lap, Z2, gcn2, b2, t, isLast,
            /*KH=*/H_, /*khShift=*/5, /*colsTotal=*/B_ * H_);
    }

    final_kernel<<<(B_ * N_ * H_ + 255) / 256, 256, 0, stream>>>(
        states, gcn1, gcn2, out);
}